// GraphConv_38543036514383
// MI455X (gfx1250) — compile-verified
//
#include <hip/hip_runtime.h>
#include <math.h>

#define N_ENT 50000
#define N_EDG 600000
#define CH 128
#define LEAKY 0.01f
#define EPSN 1e-12f

typedef __attribute__((ext_vector_type(2))) float v2f;
typedef __attribute__((ext_vector_type(8))) float v8f;

// ---------------- utility kernels ----------------

__global__ void fill_kernel(float* __restrict__ p, int n, float v) {
    int i = blockIdx.x * blockDim.x + threadIdx.x;
    if (i < n) p[i] = v;
}

__global__ void copy_kernel(const float* __restrict__ src, float* __restrict__ dst, int n) {
    int i = blockIdx.x * blockDim.x + threadIdx.x;
    if (i < n) dst[i] = src[i];
}

__global__ void count_kernel(const int* __restrict__ ei, float* __restrict__ counts) {
    int e = blockIdx.x * blockDim.x + threadIdx.x;
    if (e < N_EDG) atomicAdd(&counts[ei[e]], 1.0f);
}

// ---------------- edge scatter: agg[head] += res[tail] * weight[etype] -----
// one wave per edge; each lane owns 4 contiguous channels (float4 coalesced)

__global__ __launch_bounds__(256)
void scatter_kernel(const float* __restrict__ res,
                    const float* __restrict__ weight,
                    const int* __restrict__ ei,   // [2, E] flat, int32
                    const int* __restrict__ et,
                    float* __restrict__ agg) {
    int wave = (blockIdx.x * blockDim.x + threadIdx.x) >> 5;
    int lane = threadIdx.x & 31;
    if (wave >= N_EDG) return;
    int head = ei[wave];
    int tail = ei[N_EDG + wave];
    int rel  = et[wave];
    int c = lane * 4;
    float4 r4 = *(const float4*)(res + (size_t)tail * CH + c);
    float4 w4 = *(const float4*)(weight + (size_t)rel * CH + c);
    float* dst = agg + (size_t)head * CH + c;
    atomicAdd(dst + 0, r4.x * w4.x);
    atomicAdd(dst + 1, r4.y * w4.y);
    atomicAdd(dst + 2, r4.z * w4.z);
    atomicAdd(dst + 3, r4.w * w4.w);
}

// ---------------- scatter-mean divide + L2 row normalize -------------------
// one wave per entity row

__global__ __launch_bounds__(256)
void normalize_kernel(float* __restrict__ agg, const float* __restrict__ counts) {
    int row  = (blockIdx.x * blockDim.x + threadIdx.x) >> 5;
    int lane = threadIdx.x & 31;
    if (row >= N_ENT) return;
    float inv = 1.0f / fmaxf(counts[row], 1.0f);
    float4 v = *(float4*)(agg + (size_t)row * CH + lane * 4);
    v.x *= inv; v.y *= inv; v.z *= inv; v.w *= inv;
    float s = v.x * v.x + v.y * v.y + v.z * v.z + v.w * v.w;
    #pragma unroll
    for (int off = 16; off > 0; off >>= 1)
        s += __shfl_xor(s, off, 32);
    float scale = 1.0f / fmaxf(sqrtf(s), EPSN);
    v.x *= scale; v.y *= scale; v.z *= scale; v.w *= scale;
    *(float4*)(agg + (size_t)row * CH + lane * 4) = v;
}

// ---------------- fused dual-GEMM hop kernel (fp32 WMMA) -------------------
// out = leaky((res+agg) @ W1^T + b1) + leaky([res,agg] @ W2^T + b2)
// 32 rows per workgroup, 8 waves: wave = (rowSub 0..1) x (colQuad 0..3),
// each wave: 16 rows x 2 col-tiles of 16, three f32 WMMA chains per K-step.

#define ROWS 32
#define LSTR 132   // padded LDS row stride: bank-conflict-free A fragments

__global__ __launch_bounds__(256)
void gemm_hop_kernel(const float* __restrict__ xres, const float* __restrict__ xagg,
                     const float* __restrict__ W1, const float* __restrict__ B1,
                     const float* __restrict__ W2, const float* __restrict__ B2,
                     float* __restrict__ out) {
    __shared__ float lres[ROWS * LSTR];
    __shared__ float lagg[ROWS * LSTR];
    __shared__ float lsum[ROWS * LSTR];

    int row0 = blockIdx.x * ROWS;
    int t = threadIdx.x;

    // stage 32x128 panels of res and agg (+ precomputed sum) into LDS
    for (int i = t; i < ROWS * (CH / 4); i += 256) {
        int r = i / (CH / 4);
        int c = (i % (CH / 4)) * 4;
        int gr = row0 + r;
        float4 a = {0.f, 0.f, 0.f, 0.f}, b = {0.f, 0.f, 0.f, 0.f};
        if (gr < N_ENT) {
            a = *(const float4*)(xres + (size_t)gr * CH + c);
            b = *(const float4*)(xagg + (size_t)gr * CH + c);
        }
        *(float4*)(lres + r * LSTR + c) = a;
        *(float4*)(lagg + r * LSTR + c) = b;
        float4 s; s.x = a.x + b.x; s.y = a.y + b.y; s.z = a.z + b.z; s.w = a.w + b.w;
        *(float4*)(lsum + r * LSTR + c) = s;
    }
    __syncthreads();

    int wave = t >> 5, lane = t & 31;
    int rsub = wave >> 2;        // 0..1 : 16-row sub-tile
    int cq   = wave & 3;         // 0..3 : pair of 16-col tiles
    int al   = lane & 15;        // M (for A) / N (for B) within tile
    int koff = (lane >> 4) * 2;  // lanes 0-15: K=0,1 ; lanes 16-31: K=2,3

    const float* arow_res = lres + (rsub * 16 + al) * LSTR;
    const float* arow_agg = lagg + (rsub * 16 + al) * LSTR;
    const float* arow_sum = lsum + (rsub * 16 + al) * LSTR;

    for (int jt = 0; jt < 2; ++jt) {
        int j = cq * 32 + jt * 16;          // output column tile base
        int n = j + al;                     // this lane's B row (= W row)
        const float* w1p = W1 + (size_t)n * CH;        // W1[n, :]
        const float* w2p = W2 + (size_t)n * (2 * CH);  // W2[n, :]

        v8f acc1 = {};
        v8f acc2 = {};
        #pragma unroll 8
        for (int k = 0; k < CH; k += 4) {
            v2f aS = *(const v2f*)(arow_sum + k + koff);
            v2f b1 = *(const v2f*)(w1p + k + koff);
            acc1 = __builtin_amdgcn_wmma_f32_16x16x4_f32(
                false, aS, false, b1, (short)0, acc1, false, false);

            v2f aR  = *(const v2f*)(arow_res + k + koff);
            v2f b2a = *(const v2f*)(w2p + k + koff);
            acc2 = __builtin_amdgcn_wmma_f32_16x16x4_f32(
                false, aR, false, b2a, (short)0, acc2, false, false);

            v2f aA  = *(const v2f*)(arow_agg + k + koff);
            v2f b2b = *(const v2f*)(w2p + CH + k + koff);
            acc2 = __builtin_amdgcn_wmma_f32_16x16x4_f32(
                false, aA, false, b2b, (short)0, acc2, false, false);
        }

        // epilogue: C/D layout -> VGPR r holds M=r (lanes 0-15) / M=r+8 (16-31)
        int col   = j + (lane & 15);
        int rbase = row0 + rsub * 16 + ((lane >> 4) ? 8 : 0);
        float bb1 = B1[col], bb2 = B2[col];
        #pragma unroll
        for (int r = 0; r < 8; ++r) {
            int row = rbase + r;
            if (row < N_ENT) {
                float x1 = acc1[r] + bb1;
                x1 = (x1 >= 0.f) ? x1 : LEAKY * x1;
                float x2 = acc2[r] + bb2;
                x2 = (x2 >= 0.f) ? x2 : LEAKY * x2;
                out[(size_t)row * CH + col] = x1 + x2;
            }
        }
    }
}

// ---------------- host-side launch ----------------

extern "C" void kernel_launch(void* const* d_in, const int* in_sizes, int n_in,
                              void* d_out, int out_size, void* d_ws, size_t ws_size,
                              hipStream_t stream) {
    const float* emb    = (const float*)d_in[0];
    const int*   ei     = (const int*)d_in[1];   // [2, E]
    const int*   et     = (const int*)d_in[2];   // [E]
    const float* weight = (const float*)d_in[3]; // [32,128]
    const float* W1w    = (const float*)d_in[4]; // [2,128,128]
    const float* W1b    = (const float*)d_in[5]; // [2,128]
    const float* W2w    = (const float*)d_in[6]; // [2,128,256]
    const float* W2b    = (const float*)d_in[7]; // [2,128]
    float* out = (float*)d_out;

    float* agg    = (float*)d_ws;                       // N_ENT*CH floats
    float* resmid = agg + (size_t)N_ENT * CH;           // N_ENT*CH floats
    float* counts = resmid + (size_t)N_ENT * CH;        // N_ENT floats

    // in-degree counts (hop-invariant)
    fill_kernel<<<(N_ENT + 255) / 256, 256, 0, stream>>>(counts, N_ENT, 0.0f);
    count_kernel<<<(N_EDG + 255) / 256, 256, 0, stream>>>(ei, counts);

    const float* res = emb;
    for (int hop = 0; hop < 2; ++hop) {
        fill_kernel<<<(N_ENT * CH + 255) / 256, 256, 0, stream>>>(agg, N_ENT * CH, 0.0f);
        scatter_kernel<<<(N_EDG * 32 + 255) / 256, 256, 0, stream>>>(res, weight, ei, et, agg);
        normalize_kernel<<<(N_ENT * 32 + 255) / 256, 256, 0, stream>>>(agg, counts);
        float* dst = (hop == 0) ? resmid : out;
        gemm_hop_kernel<<<(N_ENT + ROWS - 1) / ROWS, 256, 0, stream>>>(
            res, agg,
            W1w + (size_t)hop * CH * CH,     W1b + (size_t)hop * CH,
            W2w + (size_t)hop * CH * 2 * CH, W2b + (size_t)hop * CH,
            dst);
        res = resmid;
    }

    // second output of the tuple: weight, appended flat after res
    copy_kernel<<<(32 * CH + 255) / 256, 256, 0, stream>>>(
        weight, out + (size_t)N_ENT * CH, 32 * CH);
}